// RippleNet_8400956031211
// MI455X (gfx1250) — compile-verified
//
#include <hip/hip_runtime.h>
#include <hip/hip_bf16.h>

typedef __attribute__((ext_vector_type(2))) float v2f;
typedef __attribute__((ext_vector_type(8))) float v8f;

#define B_SZ   1024
#define M_SZ   32
#define H_SZ   2
#define D_SZ   64
#define NREL   32
#define QCOLS  (NREL * D_SZ)   // 2048

// ---------------------------------------------------------------------------
// Phase 1: dense FP32 GEMM on the WMMA pipes.
//   Q[b, r*64 + j] = sum_i Item[b,i] * W[i, r*64+j]
// where Item[b,:] = entity_emb[edge_index[b,1]] (gathered A rows)
// and   W[i, r*64+j] = relation_emb[r*4096 + i*64 + j] (strided B columns).
// M=1024, N=2048, K=64. One 16x16 tile per wave, 16x V_WMMA_F32_16X16X4_F32.
// ---------------------------------------------------------------------------
__global__ void __launch_bounds__(256)
ripple_phase1_gemm(const int* __restrict__ edge_index,
                   const float* __restrict__ entity_emb,
                   const float* __restrict__ relation_emb,
                   float* __restrict__ Q)
{
    const int lane = threadIdx.x & 31;
    const int wave = threadIdx.x >> 5;
    const int tile = blockIdx.x * 8 + wave;    // 8192 tiles total
    const int tn   = tile & 127;               // 128 tiles along N (2048/16)
    const int tm   = tile >> 7;                // 64 tiles along M (1024/16)
    const int l16  = lane & 15;
    const int hi   = lane >> 4;                // 0: K pair {0,1}, 1: K pair {2,3}

    // A fragment source: gathered item row (16x4 f32 layout, ISA 7.12.2)
    const int   arow  = tm * 16 + l16;
    const int   ent   = edge_index[arow * 2 + 1];
    const float* aBase = entity_emb + (size_t)ent * D_SZ + hi * 2;

    // B fragment source: column c of W, rows i (4x16 f32 layout)
    const int   col   = tn * 16 + l16;
    const int   r     = col >> 6;
    const int   j     = col & 63;
    const float* bBase = relation_emb + (size_t)r * (D_SZ * D_SZ) + j
                         + (size_t)(hi * 2) * D_SZ;

    v8f acc = {};
    #pragma unroll
    for (int k0 = 0; k0 < D_SZ; k0 += 4) {
        v2f a = *(const v2f*)(aBase + k0);      // A[row, k0+hi*2 .. +1], 8B aligned
        v2f b;
        b.x = bBase[(size_t)k0 * D_SZ];         // W[k0+hi*2,   col]
        b.y = bBase[(size_t)(k0 + 1) * D_SZ];   // W[k0+hi*2+1, col]
        // 8-arg f32 pattern: (neg_a, A, neg_b, B, c_mod, C, reuse_a, reuse_b)
        acc = __builtin_amdgcn_wmma_f32_16x16x4_f32(
                  false, a, false, b, (short)0, acc, false, false);
    }

    // D layout: reg g -> row g + hi*8, col = l16
    float* qBase = Q + (size_t)(tm * 16) * QCOLS + tn * 16 + l16;
    #pragma unroll
    for (int g = 0; g < 8; ++g) {
        qBase[(size_t)(g + hi * 8) * QCOLS] = acc[g];
    }
}

// ---------------------------------------------------------------------------
// Phase 2: one wave32 per batch element; lane m handles ripple slot m.
//   s1 = Q[b, rel_m, :] . h_m      (== item^T R_rel h, via precomputed Q)
//   s2 = item . h_m
//   out[b] = sum_hop sum_m softmax_m(s1)_m * s2_m
// Wave-wide softmax with __shfl_xor over 32 lanes (wave32).
// ---------------------------------------------------------------------------
__global__ void __launch_bounds__(256)
ripple_phase2(const int* __restrict__ edge_index,
              const int* __restrict__ ripple_sets,
              const float* __restrict__ entity_emb,
              const float* __restrict__ Q,
              float* __restrict__ out)
{
    const int lane = threadIdx.x & 31;
    const int wave = threadIdx.x >> 5;
    const int b    = blockIdx.x * 8 + wave;

    const int    ent  = edge_index[b * 2 + 1];
    const float* item = entity_emb + (size_t)ent * D_SZ;
    const float* qrow = Q + (size_t)b * QCOLS;

    float acc = 0.0f;
    #pragma unroll
    for (int hop = 0; hop < H_SZ; ++hop) {
        const int base = (((hop * B_SZ) + b) * M_SZ + lane) * 2;  // (H,B,M,2)
        const int head = ripple_sets[base + 0];
        const int rel  = ripple_sets[base + 1];
        const float* h = entity_emb + (size_t)head * D_SZ;
        const float* q = qrow + rel * D_SZ;

        float s1 = 0.0f, s2 = 0.0f;
        #pragma unroll
        for (int jj = 0; jj < D_SZ; jj += 4) {
            float4 hv = *(const float4*)(h + jj);
            float4 qv = *(const float4*)(q + jj);
            float4 iv = *(const float4*)(item + jj);
            s1 += hv.x * qv.x + hv.y * qv.y + hv.z * qv.z + hv.w * qv.w;
            s2 += hv.x * iv.x + hv.y * iv.y + hv.z * iv.z + hv.w * iv.w;
        }

        // softmax over the 32 lanes (one slot per lane)
        float mx = s1;
        #pragma unroll
        for (int off = 16; off > 0; off >>= 1)
            mx = fmaxf(mx, __shfl_xor(mx, off, 32));
        float e = expf(s1 - mx);
        float esum = e;
        #pragma unroll
        for (int off = 16; off > 0; off >>= 1)
            esum += __shfl_xor(esum, off, 32);
        float contrib = (e / esum) * s2;
        #pragma unroll
        for (int off = 16; off > 0; off >>= 1)
            contrib += __shfl_xor(contrib, off, 32);
        acc += contrib;
    }

    if (lane == 0) out[b] = acc;
}

// ---------------------------------------------------------------------------
extern "C" void kernel_launch(void* const* d_in, const int* in_sizes, int n_in,
                              void* d_out, int out_size, void* d_ws, size_t ws_size,
                              hipStream_t stream)
{
    const int*   edge_index   = (const int*)d_in[0];     // (B, 2)
    const int*   ripple_sets  = (const int*)d_in[1];     // (H, B, M, 2)
    const float* entity_emb   = (const float*)d_in[2];   // (N_ENT, 64)
    const float* relation_emb = (const float*)d_in[3];   // (32, 4096)
    float*       out          = (float*)d_out;           // (B, 1)
    float*       Q            = (float*)d_ws;            // needs 8 MB scratch

    // Phase 1: 8192 tiles, 8 waves (one tile each) per 256-thread block.
    ripple_phase1_gemm<<<1024, 256, 0, stream>>>(edge_index, entity_emb,
                                                 relation_emb, Q);
    // Phase 2: 1024 waves, 8 per block.
    ripple_phase2<<<128, 256, 0, stream>>>(edge_index, ripple_sets,
                                           entity_emb, Q, out);
}